// IsolatedNodeExpertv2_58308476011152
// MI455X (gfx1250) — compile-verified
//
#include <hip/hip_runtime.h>
#include <hip/hip_bf16.h>

typedef __attribute__((ext_vector_type(2))) float v2f;
typedef __attribute__((ext_vector_type(8))) float v8f;

#ifndef BDIM
#define BDIM 256
#endif

// ---------------------------------------------------------------------------
// 1) zero the two degree accumulators (ws is poisoned by the harness)
// ---------------------------------------------------------------------------
__global__ void k_zero_f32(float* __restrict__ p, int n) {
  int i = blockIdx.x * blockDim.x + threadIdx.x;
  if (i < n) p[i] = 0.0f;
}

// ---------------------------------------------------------------------------
// 2) out/in degrees via L2-resident fp32 atomics (exact up to 2^24 counts)
//    edge_index is int64 in the reference -> long long
// ---------------------------------------------------------------------------
__global__ void k_degrees(const long long* __restrict__ ei,
                          float* __restrict__ deg_src,
                          float* __restrict__ deg_dst,
                          long long E) {
  long long e = (long long)blockIdx.x * blockDim.x + threadIdx.x;
  if (e >= E) return;
  long long s = ei[e];
  long long d = ei[E + e];
  atomicAdd(&deg_src[s], 1.0f);
  atomicAdd(&deg_dst[d], 1.0f);
}

// ---------------------------------------------------------------------------
// 3) per-node normalizers: dinv = rsqrt(deg_dst + 1), iso = 1/(deg_src+1e-3)
// ---------------------------------------------------------------------------
__global__ void k_nodeprep(const float* __restrict__ deg_src,
                           const float* __restrict__ deg_dst,
                           float* __restrict__ dinv,
                           float* __restrict__ iso, int N) {
  int n = blockIdx.x * blockDim.x + threadIdx.x;
  if (n >= N) return;
  dinv[n] = rsqrtf(deg_dst[n] + 1.0f);
  iso[n]  = 1.0f / (deg_src[n] + 0.001f);
}

// ---------------------------------------------------------------------------
// 4) x_w[b,n] = mean_t x[b,t,n] * iso[n]; seed agg with self-loop term
//    (adjacent threads -> adjacent n: every load of the 102 MB x is coalesced)
// ---------------------------------------------------------------------------
__global__ void k_xmean(const float* __restrict__ x,
                        const float* __restrict__ iso,
                        const float* __restrict__ dinv,
                        float* __restrict__ xw,
                        float* __restrict__ agg,
                        int N, int B, int T) {
  long long idx = (long long)blockIdx.x * blockDim.x + threadIdx.x; // b*N + n
  if (idx >= (long long)B * N) return;
  int b = (int)(idx / N);
  int n = (int)(idx % N);
  const float* p = x + (size_t)b * T * N + n;
  float s = 0.0f;
#pragma unroll 8
  for (int t = 0; t < T; ++t) s += p[(size_t)t * N];
  float w = (s / (float)T) * iso[n];
  xw[idx] = w;
  float di = dinv[n];
  agg[idx] = w * di * di;            // norm_ii = dinv_i^2 self-loop term
}

// ---------------------------------------------------------------------------
// 5) edge scatter: agg[b,dst] += xw[b,src] * dinv[src]*dinv[dst]
//    agg is 1.6 MB -> lives in L2 (192 MB); atomics resolve there
// ---------------------------------------------------------------------------
__global__ void k_scatter(const long long* __restrict__ ei,
                          const float* __restrict__ dinv,
                          const float* __restrict__ xw,
                          float* __restrict__ agg,
                          int N, long long E, int B) {
  long long e = (long long)blockIdx.x * blockDim.x + threadIdx.x;
  if (e >= E) return;
  long long s = ei[e];
  long long d = ei[E + e];
  float nrm = dinv[s] * dinv[d];
#pragma unroll
  for (int b = 0; b < 8; ++b) {
    if (b >= B) break;
    atomicAdd(&agg[(size_t)b * N + d], xw[(size_t)b * N + s] * nrm);
  }
}

// ---------------------------------------------------------------------------
// 6) the surviving matmul: [2,H] @ [H,HOR] with fp32 WMMA 16x16x4.
//    A row 0 = W_gcn[0], row 1 = b_gcn, rows 2..15 = 0.
//    D[0,o] = v[o], D[1,o] + b_proj[o] = c[o].
//    One wave, two N-tiles (HOR=24 -> ceil(24/16)=2), K stepped by 4.
//    ISA layouts (CDNA5 ISA):
//      A 16x4: lane<16 VGPR0=K0,VGPR1=K1 ; lane>=16 VGPR0=K2,VGPR1=K3
//      B 4x16: same K striping, N = lane&15
//      D: VGPR j holds M=j (lanes 0-15), so M=0 -> acc[0], M=1 -> acc[1]
// ---------------------------------------------------------------------------
__global__ void k_wmma_vc(const float* __restrict__ W_gcn,   // [H]
                          const float* __restrict__ b_gcn,   // [H]
                          const float* __restrict__ W_proj,  // [H, HOR] row-major
                          const float* __restrict__ b_proj,  // [HOR]
                          float* __restrict__ v_out,         // [>=HOR]
                          float* __restrict__ c_out,         // [>=HOR]
                          int H, int HOR) {
  const int lane   = threadIdx.x;       // 0..31, one wave32
  const int lane16 = lane & 15;
  const int half   = lane >> 4;         // selects K pair {0,1} vs {2,3}
  const int ntiles = (HOR + 15) / 16;

  for (int t = 0; t < ntiles; ++t) {
    const int o = t * 16 + lane16;
    v8f acc = {};
    for (int kk = 0; kk < H; kk += 4) {
      const int k0 = kk + 2 * half;
      // A operand: only rows M=0 (W_gcn) and M=1 (b_gcn) are nonzero.
      float a0 = 0.0f, a1 = 0.0f;
      if (lane16 == 0) { a0 = W_gcn[k0]; a1 = W_gcn[k0 + 1]; }
      if (lane16 == 1) { a0 = b_gcn[k0]; a1 = b_gcn[k0 + 1]; }
      // B operand: W_proj[k, o] with zero-padding past HOR.
      float b0 = 0.0f, b1 = 0.0f;
      if (o < HOR) {
        b0 = W_proj[(size_t)k0 * HOR + o];
        b1 = W_proj[(size_t)(k0 + 1) * HOR + o];
      }
      v2f A = {a0, a1};
      v2f Bv = {b0, b1};
      // 8 args: (neg_a, A, neg_b, B, c_mod, C, reuse_a, reuse_b)
      acc = __builtin_amdgcn_wmma_f32_16x16x4_f32(
          false, A, false, Bv, (short)0, acc, false, false);
    }
    if (lane < 16 && o < HOR) {
      v_out[o] = acc[0];                 // row M=0
      c_out[o] = acc[1] + b_proj[o];     // row M=1 + projection bias
    }
  }
}

// ---------------------------------------------------------------------------
// 7) epilogue: out[b,n,o] = agg[b,n]*v[o] + c[o], float4-vectorized (HOR%4==0)
// ---------------------------------------------------------------------------
__global__ void k_output(const float* __restrict__ agg,
                         const float* __restrict__ v,
                         const float* __restrict__ c,
                         float4* __restrict__ out4,
                         long long BN, int HOR4) {
  long long idx = (long long)blockIdx.x * blockDim.x + threadIdx.x;
  if (idx >= BN * HOR4) return;
  int q = (int)(idx % HOR4);
  long long bn = idx / HOR4;
  float a = agg[bn];
  const float4 vq = ((const float4*)v)[q];
  const float4 cq = ((const float4*)c)[q];
  float4 r;
  r.x = fmaf(a, vq.x, cq.x);
  r.y = fmaf(a, vq.y, cq.y);
  r.z = fmaf(a, vq.z, cq.z);
  r.w = fmaf(a, vq.w, cq.w);
  out4[idx] = r;
}

// ---------------------------------------------------------------------------
extern "C" void kernel_launch(void* const* d_in, const int* in_sizes, int n_in,
                              void* d_out, int out_size, void* d_ws, size_t ws_size,
                              hipStream_t stream) {
  const float*     x      = (const float*)d_in[0];      // [B,T,N,1]
  const long long* ei     = (const long long*)d_in[1];  // [2,E] int64
  const float*     W_gcn  = (const float*)d_in[2];      // [1,H]
  const float*     b_gcn  = (const float*)d_in[3];      // [H]
  const float*     W_proj = (const float*)d_in[4];      // [H,HOR]
  const float*     b_proj = (const float*)d_in[5];      // [HOR]
  float*           out    = (float*)d_out;              // [B,N,HOR]

  const int B = 8, T = 64;
  const int H   = in_sizes[2];                 // 128
  const int HOR = in_sizes[5];                 // 24
  const int N   = in_sizes[0] / (B * T);       // 50000
  const long long E = (long long)in_sizes[1] / 2;  // 1,600,000
  const long long BN = (long long)B * N;

  // workspace layout (floats)
  float* ws      = (float*)d_ws;
  float* deg_src = ws;                 // [N]
  float* deg_dst = deg_src + N;        // [N]
  float* dinv    = deg_dst + N;        // [N]
  float* iso     = dinv + N;           // [N]
  float* xw      = iso + N;            // [B*N]
  float* agg     = xw + BN;            // [B*N]
  float* v_buf   = agg + BN;           // [32]
  float* c_buf   = v_buf + 32;         // [32]

  auto cdiv = [](long long a, long long b) { return (int)((a + b - 1) / b); };

  k_zero_f32 <<<cdiv(2LL * N, BDIM), BDIM, 0, stream>>>(deg_src, 2 * N);
  k_degrees  <<<cdiv(E, BDIM),       BDIM, 0, stream>>>(ei, deg_src, deg_dst, E);
  k_nodeprep <<<cdiv(N, BDIM),       BDIM, 0, stream>>>(deg_src, deg_dst, dinv, iso, N);
  k_xmean    <<<cdiv(BN, BDIM),      BDIM, 0, stream>>>(x, iso, dinv, xw, agg, N, B, T);
  k_scatter  <<<cdiv(E, BDIM),       BDIM, 0, stream>>>(ei, dinv, xw, agg, N, E, B);
  k_wmma_vc  <<<1, 32, 0, stream>>>(W_gcn, b_gcn, W_proj, b_proj, v_buf, c_buf, H, HOR);
  k_output   <<<cdiv(BN * (HOR / 4), BDIM), BDIM, 0, stream>>>(agg, v_buf, c_buf,
                                                               (float4*)out, BN, HOR / 4);
}